// GCNLayer_20547123544254
// MI455X (gfx1250) — compile-verified
//
#include <hip/hip_runtime.h>
#include <stdint.h>

typedef __attribute__((ext_vector_type(2))) float v2f;
typedef __attribute__((ext_vector_type(8))) float v8f;

#define N_NODES   50000
#define N_EDGES   800000
#define D_IN      256
#define D_OUT     64
#define NEG_SLOPE 0.2f

#define ROWS_PER_BLOCK 128   // 8 waves * 16 rows
#define KC             64    // K-chunk staged in LDS
#define LDS_PAD        4     // keeps 16B alignment and kills bank conflicts

// Issue one per-lane async global->LDS 16B copy (ASYNCcnt-tracked, no VGPR dest).
__device__ __forceinline__ void async_copy_b128(const float* gptr, float* lptr) {
    const uint64_t ga = (uint64_t)(uintptr_t)gptr;
    // generic pointer to __shared__ = {aperture_hi32, lds_offset_lo32}
    const uint32_t la = (uint32_t)(uintptr_t)lptr;
    asm volatile("global_load_async_to_lds_b128 %0, %1, off"
                 :: "v"(la), "v"(ga) : "memory");
}

__device__ __forceinline__ void wait_async_all() {
    asm volatile("s_wait_asynccnt 0x0" ::: "memory");
}

// ---------------------------------------------------------------------------
// Kernel 1: support = x @ W^T + b   (f32 WMMA 16x16x4, exact f32 precision)
// ---------------------------------------------------------------------------
__global__ __launch_bounds__(256) void gcn_gemm_wmma(
    const float* __restrict__ x,      // [N, 256]
    const float* __restrict__ W,      // [64, 256]
    const float* __restrict__ bias,   // [64]
    float* __restrict__ support,      // [N, 64]
    int n_rows)
{
    __shared__ float xlds[ROWS_PER_BLOCK][KC + LDS_PAD]; // 128*68*4 = 34816 B
    __shared__ float wlds[D_OUT][KC + LDS_PAD];          //  64*68*4 = 17408 B

    const int tid   = threadIdx.x;
    const int wave  = tid >> 5;        // 0..7
    const int lane  = tid & 31;
    const int l16   = lane & 15;
    const int khalf = lane >> 4;       // 0 or 1
    const int rowbase = blockIdx.x * ROWS_PER_BLOCK;

    // Accumulators: 4 column tiles of 16 (covers D_OUT=64), init with bias.
    // C layout: vgpr j -> row (j + 8*khalf), col = l16 (+16*t).
    v8f acc[4];
    #pragma unroll
    for (int t = 0; t < 4; ++t) {
        const float bt = bias[t * 16 + l16];
        #pragma unroll
        for (int j = 0; j < 8; ++j) acc[t][j] = bt;
    }

    const int r0 = tid >> 4;   // 0..15 : row within 16-row load group
    const int q  = tid & 15;   // 0..15 : float4 quad within 64-float K-chunk

    for (int kc = 0; kc < D_IN; kc += KC) {
        // ---- async stage: x tile [128 x 64], 8 x b128 per thread, all in flight
        #pragma unroll
        for (int rr = 0; rr < ROWS_PER_BLOCK; rr += 16) {
            int row = rowbase + rr + r0;
            int rc  = row < n_rows ? row : (n_rows - 1);   // clamp, store guarded
            async_copy_b128(&x[(size_t)rc * D_IN + kc + q * 4],
                            &xlds[rr + r0][q * 4]);
        }
        // ---- async stage: W tile [64 x 64], 4 x b128 per thread ----
        #pragma unroll
        for (int rr = 0; rr < D_OUT; rr += 16) {
            async_copy_b128(&W[(size_t)(rr + r0) * D_IN + kc + q * 4],
                            &wlds[rr + r0][q * 4]);
        }
        wait_async_all();      // all async LDS writes complete
        __syncthreads();       // ... and visible to every wave

        // ---- WMMA over the K-chunk: 16 steps of K=4, 4 column tiles ----
        const int arow = wave * 16 + l16;
        #pragma unroll
        for (int k = 0; k < KC; k += 4) {
            const int kk = k + 2 * khalf;
            v2f a;
            a.x = xlds[arow][kk];
            a.y = xlds[arow][kk + 1];
            #pragma unroll
            for (int t = 0; t < 4; ++t) {
                v2f bb;
                bb.x = wlds[t * 16 + l16][kk];
                bb.y = wlds[t * 16 + l16][kk + 1];
                // (neg_a, A, neg_b, B, c_mod, C, reuse_a, reuse_b)
                acc[t] = __builtin_amdgcn_wmma_f32_16x16x4_f32(
                    false, a, false, bb, (short)0, acc[t], false, false);
            }
        }
        __syncthreads();       // protect LDS before next chunk overwrites
    }

    // ---- store D: row = rowbase + wave*16 + j + 8*khalf, col = t*16 + l16 ----
    #pragma unroll
    for (int t = 0; t < 4; ++t) {
        #pragma unroll
        for (int j = 0; j < 8; ++j) {
            const int row = rowbase + wave * 16 + j + 8 * khalf;
            if (row < n_rows)
                support[(size_t)row * D_OUT + t * 16 + l16] = acc[t][j];
        }
    }
}

// ---------------------------------------------------------------------------
// Kernel 2: segment-sum over sorted edge_dst (binary search, no atomics)
//           + fused leaky-ReLU.  64 threads per node, 4 nodes per block.
// ---------------------------------------------------------------------------
__global__ __launch_bounds__(256) void gcn_agg(
    const float* __restrict__ support,   // [N, 64]
    const int*   __restrict__ esrc,      // [E]
    const int*   __restrict__ edst,      // [E] sorted
    const float* __restrict__ eval,      // [E]
    float* __restrict__ out)             // [N, 64]
{
    const int node = blockIdx.x * 4 + (threadIdx.x >> 6);
    const int f    = threadIdx.x & 63;
    if (node >= N_NODES) return;

    // lower_bound(node) and lower_bound(node+1) in sorted edst
    int lo, hi;
    {
        int l = 0, r = N_EDGES;
        while (l < r) { int m = (l + r) >> 1; if (edst[m] < node) l = m + 1; else r = m; }
        lo = l;
        r = N_EDGES;
        while (l < r) { int m = (l + r) >> 1; if (edst[m] <= node) l = m + 1; else r = m; }
        hi = l;
    }

    // 2-way accumulation for ILP; gathers are 256B coalesced per edge,
    // support (12.8 MB) is L2-resident on MI455X (192 MB L2).
    float acc0 = 0.f, acc1 = 0.f;
    int e = lo;
    for (; e + 1 < hi; e += 2) {
        acc0 = fmaf(eval[e],     support[(size_t)esrc[e]     * D_OUT + f], acc0);
        acc1 = fmaf(eval[e + 1], support[(size_t)esrc[e + 1] * D_OUT + f], acc1);
    }
    if (e < hi)
        acc0 = fmaf(eval[e], support[(size_t)esrc[e] * D_OUT + f], acc0);

    const float acc = acc0 + acc1;
    out[(size_t)node * D_OUT + f] = acc >= 0.f ? acc : NEG_SLOPE * acc;
}

// ---------------------------------------------------------------------------
extern "C" void kernel_launch(void* const* d_in, const int* in_sizes, int n_in,
                              void* d_out, int out_size, void* d_ws, size_t ws_size,
                              hipStream_t stream) {
    const float* x    = (const float*)d_in[0];
    const float* W    = (const float*)d_in[1];
    const float* bias = (const float*)d_in[2];
    const int*   esrc = (const int*)d_in[3];
    const int*   edst = (const int*)d_in[4];
    const float* eval = (const float*)d_in[5];
    float* out = (float*)d_out;

    float* support = (float*)d_ws;   // 50000*64*4 = 12.8 MB scratch

    const int gemm_blocks = (N_NODES + ROWS_PER_BLOCK - 1) / ROWS_PER_BLOCK;
    gcn_gemm_wmma<<<gemm_blocks, 256, 0, stream>>>(x, W, bias, support, N_NODES);

    const int agg_blocks = (N_NODES + 3) / 4;
    gcn_agg<<<agg_blocks, 256, 0, stream>>>(support, esrc, edst, eval, out);
}